// DetPostProcessor_20169166422043
// MI455X (gfx1250) — compile-verified
//
#include <hip/hip_runtime.h>
#include <hip/hip_bf16.h>
#include <stdint.h>

#define KSEL     300
#define TPB      512
#define VPT      4
#define TILE     (TPB * VPT)      // 2048 elements per tile
#define NBUF     4                // async pipeline depth (3 outstanding/wave)
#define NBINS    4096
#define CHUNK    (NBINS / TPB)    // 8 bins per thread
#define CAND_MAX 1024

// ---- order-preserving float <-> uint32 key (bigger float => bigger key) ----
__device__ __forceinline__ uint32_t fkey(float f) {
    uint32_t u = __float_as_uint(f);
    return (u & 0x80000000u) ? ~u : (u | 0x80000000u);
}
__device__ __forceinline__ float key_to_float(uint32_t k) {
    uint32_t u = (k & 0x80000000u) ? (k & 0x7FFFFFFFu) : ~k;
    return __uint_as_float(u);
}

// ---- CDNA5 async global->LDS copy (tracked by ASYNCcnt) ----
// vdst = VGPR with LDS byte offset (ISA 10.2);
// vaddr = 64-bit global address; saddr = off (GV mode).
__device__ __forceinline__ void async_ld16_to_lds(uint32_t lds_off, const float* g) {
    asm volatile("global_load_async_to_lds_b128 %0, %1, off"
                 :: "v"(lds_off), "v"((unsigned long long)(uintptr_t)g)
                 : "memory");
}
// Wait until at most `rem` of this wave's async ops are outstanding.
__device__ __forceinline__ void wait_async_le(int rem) {
    switch (rem) {
        case 0:  asm volatile("s_wait_asynccnt 0x0" ::: "memory"); break;
        case 1:  asm volatile("s_wait_asynccnt 0x1" ::: "memory"); break;
        case 2:  asm volatile("s_wait_asynccnt 0x2" ::: "memory"); break;
        default: asm volatile("s_wait_asynccnt 0x3" ::: "memory"); break;
    }
}

// Stream all tiles of one batch through a NBUF-deep async global->LDS pipeline.
// Each thread consumes exactly the 16B it loaded, so only per-wave ASYNCcnt
// waits are required (async loads complete in order within a wave).
template <typename Consume>
__device__ __forceinline__ void stream_tiles(const float* __restrict__ base, int N,
                                             int tid, const uint32_t* ldsOff,
                                             float (*buf)[TILE], Consume&& consume)
{
    const int ntiles = (N + TILE - 1) / TILE;
    const int myOff  = tid * VPT;
    // last tile index in which this thread participates (myOff < N always here)
    const int lastT  = (N - 1 - myOff) / TILE;

    // prologue: fill the pipeline
    for (int t = 0; t < NBUF - 1 && t < ntiles; ++t) {
        const int e = t * TILE + myOff;
        if (e < N) async_ld16_to_lds(ldsOff[t & (NBUF - 1)], base + e);
    }
    for (int t = 0; t < ntiles; ++t) {
        const int ti = t + NBUF - 1;
        if (ti < ntiles) {
            const int e = ti * TILE + myOff;
            if (e < N) async_ld16_to_lds(ldsOff[ti & (NBUF - 1)], base + e);
        }
        // force completion of *this thread's* tile-t load: after it completes,
        // at most (lastT - t) of our loads (tiles t+1..lastT) remain in flight.
        int rem = lastT - t;
        if (rem > NBUF - 1) rem = NBUF - 1;
        if (rem < 0) rem = 0;
        wait_async_le(rem);

        const int e = t * TILE + myOff;
        if (e < N) {
            const float4 v = *reinterpret_cast<const float4*>(&buf[t & (NBUF - 1)][myOff]);
            consume(v, e);
        }
    }
}

extern "C" __global__ __launch_bounds__(TPB)
void detr_topk_kernel(const float* __restrict__ logits,
                      const float* __restrict__ boxes,
                      const float* __restrict__ tsizes,
                      float* __restrict__ outScores,
                      float* __restrict__ outLabels,
                      float* __restrict__ outBoxes,
                      int Q, int C)
{
    const int b   = blockIdx.x;
    const int N   = Q * C;                       // 230400
    const int tid = threadIdx.x;
    const float* base = logits + (size_t)b * (size_t)N;

    __shared__ uint32_t hist[NBINS];
    __shared__ uint32_t chunkSum[TPB];
    __shared__ __align__(16) float buf[NBUF][TILE];
    __shared__ uint32_t candKey[CAND_MAX];
    __shared__ uint32_t candIdx[CAND_MAX];
    __shared__ uint32_t selKey[KSEL];
    __shared__ uint32_t selIdx[KSEL];
    __shared__ uint32_t s_thresh, s_cntgt, s_selCount, s_candCount;

    for (int i = tid; i < NBINS; i += TPB) hist[i] = 0u;
    if (tid == 0) { s_selCount = 0u; s_candCount = 0u; }
    __syncthreads();

    uint32_t ldsOff[NBUF];
#pragma unroll
    for (int k = 0; k < NBUF; ++k)
        ldsOff[k] = (uint32_t)(uintptr_t)&buf[k][tid * VPT];

    // ================= PASS 1: histogram of key top-12 bits =================
    stream_tiles(base, N, tid, ldsOff, buf, [&](const float4& v, int /*e*/) {
        const float vv[VPT] = { v.x, v.y, v.z, v.w };
#pragma unroll
        for (int j = 0; j < VPT; ++j) {
            const uint32_t bin = fkey(vv[j]) >> 20;
            atomicAdd(&hist[bin], 1u);
        }
    });
    __syncthreads();

    // --- find threshold bin T: count(bins > T) < K <= count(bins >= T) ---
    {
        uint32_t s = 0;
#pragma unroll
        for (int j = 0; j < CHUNK; ++j) s += hist[tid * CHUNK + j];
        chunkSum[tid] = s;
    }
    __syncthreads();
    if (tid == 0) {
        uint32_t acc = 0, cgt = 0;
        int T = 0;
        bool found = false;
        for (int c = TPB - 1; c >= 0 && !found; --c) {
            if (acc + chunkSum[c] >= (uint32_t)KSEL) {
                for (int bn = c * CHUNK + CHUNK - 1; bn >= c * CHUNK; --bn) {
                    if (acc + hist[bn] >= (uint32_t)KSEL) { T = bn; cgt = acc; found = true; break; }
                    acc += hist[bn];
                }
            } else {
                acc += chunkSum[c];
            }
        }
        s_thresh = (uint32_t)T;
        s_cntgt  = cgt;
    }
    __syncthreads();
    const uint32_t T = s_thresh;

    // ========== PASS 2: collect winners + threshold-bin candidates ==========
    stream_tiles(base, N, tid, ldsOff, buf, [&](const float4& v, int e) {
        const float vv[VPT] = { v.x, v.y, v.z, v.w };
#pragma unroll
        for (int j = 0; j < VPT; ++j) {
            const uint32_t k   = fkey(vv[j]);
            const uint32_t bin = k >> 20;
            if (bin > T) {
                const uint32_t pos = atomicAdd(&s_selCount, 1u);
                if (pos < (uint32_t)KSEL) { selKey[pos] = k; selIdx[pos] = (uint32_t)(e + j); }
            } else if (bin == T) {
                const uint32_t pos = atomicAdd(&s_candCount, 1u);
                if (pos < (uint32_t)CAND_MAX) { candKey[pos] = k; candIdx[pos] = (uint32_t)(e + j); }
            }
        }
    });
    __syncthreads();

    // --- rank candidates (ties: lower index first), take remaining slots ---
    const uint32_t cgt    = s_cntgt;
    const uint32_t needed = (uint32_t)KSEL - cgt;
    uint32_t Ccnt = s_candCount;
    if (Ccnt > (uint32_t)CAND_MAX) Ccnt = (uint32_t)CAND_MAX;

    for (uint32_t i = tid; i < Ccnt; i += TPB) {
        const uint32_t ki = candKey[i], xi = candIdx[i];
        uint32_t rank = 0;
        for (uint32_t j = 0; j < Ccnt; ++j) {
            const uint32_t kj = candKey[j];
            rank += (kj > ki) || (kj == ki && candIdx[j] < xi);
        }
        if (rank < needed) {
            const uint32_t pos = atomicAdd(&s_selCount, 1u);
            if (pos < (uint32_t)KSEL) { selKey[pos] = ki; selIdx[pos] = xi; }
        }
    }
    __syncthreads();

    // --- final descending rank-sort of the <=300 winners and emit ---
    uint32_t M = s_selCount;
    if (M > (uint32_t)KSEL) M = (uint32_t)KSEL;
    const float img_h = tsizes[b * 2 + 0];
    const float img_w = tsizes[b * 2 + 1];

    for (uint32_t i = tid; i < M; i += TPB) {
        const uint32_t ki = selKey[i], xi = selIdx[i];
        uint32_t r = 0;
        for (uint32_t j = 0; j < M; ++j) {
            const uint32_t kj = selKey[j];
            r += (kj > ki) || (kj == ki && selIdx[j] < xi);
        }
        const float x     = key_to_float(ki);
        const float score = 1.0f / (1.0f + __expf(-x));
        const uint32_t label = xi % (uint32_t)C;
        const uint32_t q     = xi / (uint32_t)C;
        const float* bp = boxes + ((size_t)b * (size_t)Q + q) * 4;
        const float cx = bp[0], cy = bp[1], w = bp[2], h = bp[3];
        const float x0 = cx - 0.5f * w, y0 = cy - 0.5f * h;
        const float x1 = cx + 0.5f * w, y1 = cy + 0.5f * h;
        const size_t o = (size_t)b * KSEL + r;
        outScores[o] = score;
        outLabels[o] = (float)label;
        float* ob = outBoxes + o * 4;
        ob[0] = x0 * img_w; ob[1] = y0 * img_h;
        ob[2] = x1 * img_w; ob[3] = y1 * img_h;
    }
    // deterministic fill of any unfilled tail slots (degenerate tie overflow)
    for (uint32_t r = M + tid; r < (uint32_t)KSEL; r += TPB) {
        const size_t o = (size_t)b * KSEL + r;
        outScores[o] = 0.0f;
        outLabels[o] = 0.0f;
        float* ob = outBoxes + o * 4;
        ob[0] = 0.0f; ob[1] = 0.0f; ob[2] = 0.0f; ob[3] = 0.0f;
    }
}

extern "C" void kernel_launch(void* const* d_in, const int* in_sizes, int n_in,
                              void* d_out, int out_size, void* d_ws, size_t ws_size,
                              hipStream_t stream) {
    (void)n_in; (void)out_size; (void)d_ws; (void)ws_size;
    const float* logits = (const float*)d_in[0];   // (B, Q, C) f32
    const float* boxes  = (const float*)d_in[1];   // (B, Q, 4) f32
    const float* tsz    = (const float*)d_in[2];   // (B, 2)    f32

    const int B = in_sizes[2] / 2;
    const int Q = in_sizes[1] / (B * 4);
    const int C = in_sizes[0] / (B * Q);

    float* out       = (float*)d_out;
    float* outScores = out;                              // B*K
    float* outLabels = out + (size_t)B * KSEL;           // B*K
    float* outBoxes  = out + (size_t)2 * B * KSEL;       // B*K*4

    detr_topk_kernel<<<dim3(B), dim3(TPB), 0, stream>>>(
        logits, boxes, tsz, outScores, outLabels, outBoxes, Q, C);
}